// MixtureOfExperts2dRouter_15599321219671
// MI455X (gfx1250) — compile-verified
//
#include <hip/hip_runtime.h>
#include <math.h>

namespace {
constexpr int kC    = 16;
constexpr int kHW   = 128 * 128;        // 16384
constexpr int kPix  = 16 * kHW;         // B*H*W = 262144
constexpr int kElem = kC * kPix;        // 4194304 elements per output tensor
constexpr int kBlk  = 256;              // 8 wave32s per block
}

// ---- CDNA5 async global->LDS staging (ASYNCcnt path) ----------------------
typedef __attribute__((address_space(1))) int g1i_t;   // global (AS1) int
typedef __attribute__((address_space(3))) int l3i_t;   // LDS (AS3) int

__device__ __forceinline__ void async_ld_b32(const void* gsrc, void* ldst) {
#if __has_builtin(__builtin_amdgcn_global_load_async_to_lds_b32)
  __builtin_amdgcn_global_load_async_to_lds_b32((g1i_t*)gsrc, (l3i_t*)ldst, 0, 0);
#else
  unsigned loff = (unsigned)(unsigned long long)(l3i_t*)ldst;  // LDS byte offset
  asm volatile("global_load_async_to_lds_b32 %0, %1, off"
               :: "v"(loff), "v"(gsrc) : "memory");
#endif
}

__device__ __forceinline__ void wait_async0() {
#if __has_builtin(__builtin_amdgcn_s_wait_asynccnt)
  __builtin_amdgcn_s_wait_asynccnt(0);
#else
  asm volatile("s_wait_asynccnt 0" ::: "memory");
#endif
}

// softplus(t) = max(t,0) + log(1 + exp(-|t|)); trans-unit version (v_exp+v_log)
__device__ __forceinline__ float softplus_f(float t) {
  return fmaxf(t, 0.0f) + __logf(1.0f + __expf(-fabsf(t)));
}

// erf via Abramowitz-Stegun 7.1.26: max abs error 1.5e-7.
// ~10 VALU + v_rcp + v_exp, vs ~25 VALU for the library erff.
// x=+/-inf -> t=0, exp(-inf)=0 -> +/-1 (matches erff saturation).
__device__ __forceinline__ float erf_fast(float v) {
  const float a1 = 0.254829592f, a2 = -0.284496736f, a3 = 1.421413741f;
  const float a4 = -1.453152027f, a5 = 1.061405429f, p = 0.3275911f;
  const float ax = fabsf(v);
  const float t  = __builtin_amdgcn_rcpf(fmaf(p, ax, 1.0f));
  float poly = fmaf(a5, t, a4);
  poly = fmaf(poly, t, a3);
  poly = fmaf(poly, t, a2);
  poly = fmaf(poly, t, a1);
  poly *= t;
  const float r = 1.0f - poly * __expf(-ax * ax);
  return copysignf(r, v);
}

__global__ __launch_bounds__(kBlk) void moe2d_router_kernel(
    const float* __restrict__ x, const float* __restrict__ noise,
    const float* __restrict__ wgp, const float* __restrict__ wnp,
    float* __restrict__ outG, float* __restrict__ outL) {
  __shared__ float xbuf[kC][kBlk];
  __shared__ float nbuf[kC][kBlk];

  const int tid = threadIdx.x;
  const int bi  = blockIdx.x;
  const int b   = bi >> 6;                 // 64 blocks per batch image
  const int s   = ((bi & 63) << 8) | tid;  // spatial offset in [0, 16384)
  const long long base = (long long)b * (kC * kHW) + s;

  // Kick off all 32 async copies (x,noise for 16 channels); each lane moves
  // 4B into its private LDS slot — fully coalesced 128B/wave transactions,
  // zero destination-VGPR pressure while in flight.
#pragma unroll
  for (int c = 0; c < kC; ++c) {
    async_ld_b32(&x[base + c * kHW],     &xbuf[c][tid]);
    async_ld_b32(&noise[base + c * kHW], &nbuf[c][tid]);
  }

  // Uniform per-channel params -> s_load_b512 pair (confirmed in asm).
  float wgp_r[kC], wnp_r[kC];
#pragma unroll
  for (int c = 0; c < kC; ++c) { wgp_r[c] = wgp[c]; wnp_r[c] = wnp[c]; }

  wait_async0();

  float wg[kC], wn[kC];
  float m1 = -INFINITY, m2 = -INFINITY, ssum = 0.0f;
  int   amax = 0;

#pragma unroll
  for (int c = 0; c < kC; ++c) {
    const float xv = xbuf[c][tid];
    const float nv = nbuf[c][tid];
    const float g  = xv * wgp_r[c];
    const float w  = softplus_f(xv * wnp_r[c]);
    const float H  = fmaf(nv, w, g);
    wg[c] = g;
    wn[c] = w;
    // online top-2 + rescaled softmax sum (sum of exp(H - running max))
    if (H > m1) {               // strict > keeps first-occurrence argmax
      m2   = m1;
      ssum = ssum * __expf(m1 - H) + 1.0f;   // first iter: 0*exp(-inf)+1 = 1
      m1   = H;
      amax = c;
    } else {
      m2 = fmaxf(m2, H);
      ssum += __expf(H - m1);
    }
  }

  const float ginv = 1.0f / ssum;  // softmax value at the argmax channel

#pragma unroll
  for (int c = 0; c < kC; ++c) {
    const bool  top  = (c == amax);
    const float excl = top ? m2 : m1;
    const float ll   = erf_fast(__fdividef(wg[c] - excl, wn[c]));
    const float Gc   = top ? ginv : 0.0f;
    const long long o = base + (long long)c * kHW;
    __builtin_nontemporal_store(Gc, &outG[o]);  // single-use: bypass cache rinse
    __builtin_nontemporal_store(ll, &outL[o]);
  }
}

extern "C" void kernel_launch(void* const* d_in, const int* in_sizes, int n_in,
                              void* d_out, int out_size, void* d_ws, size_t ws_size,
                              hipStream_t stream) {
  (void)in_sizes; (void)n_in; (void)d_ws; (void)ws_size; (void)out_size;
  const float* x     = (const float*)d_in[0];
  const float* noise = (const float*)d_in[1];
  const float* wgp   = (const float*)d_in[2];
  const float* wnp   = (const float*)d_in[3];
  float* outG = (float*)d_out;          // G first, then load_loss, flat
  float* outL = outG + kElem;
  moe2d_router_kernel<<<dim3(kPix / kBlk), dim3(kBlk), 0, stream>>>(
      x, noise, wgp, wnp, outG, outL);
}